// ChannelKillerOriginal_54365696033604
// MI455X (gfx1250) — compile-verified
//
#include <hip/hip_runtime.h>

// ChannelKiller: out[b,c,t] = x[b,c,t] * (c==0 ? 1.0 : 0.5)
// Shapes: B=16, C=8, T=262144 (= 2^18), f32.
// 128 MiB in + 128 MiB out, 1 FLOP/elem -> pure HBM-bandwidth problem
// (~11 us floor at 23.3 TB/s). WMMA intentionally unused (AI = 0.125 FLOP/B).

typedef __attribute__((ext_vector_type(4))) float v4f;

// For float4 index i: element index = 4*i, channel = (4*i >> 18) & 7 = (i >> 16) & 7.
__device__ __forceinline__ float chan_scale(unsigned i4) {
    return (((i4 >> 16) & 7u) == 0u) ? 1.0f : 0.5f;
}

// Fast path: grid exactly covers n4 (n4 == 4 * span). No bounds checks ->
// straight-line code: 4 back-to-back b128 loads, 4 NT b128 stores.
__global__ __launch_bounds__(256) void
channel_scale_b128_exact(const v4f* __restrict__ x, v4f* __restrict__ out,
                         unsigned span) {
    const unsigned tid = blockIdx.x * blockDim.x + threadIdx.x;

    const unsigned i0 = tid;
    const unsigned i1 = tid + span;
    const unsigned i2 = tid + 2u * span;
    const unsigned i3 = tid + 3u * span;

    // 4 loads in flight before any wait; RT policy keeps the input
    // resident in the 192 MB L2 across graph replays.
    v4f a = x[i0];
    v4f b = x[i1];
    v4f c = x[i2];
    v4f d = x[i3];

    a *= chan_scale(i0);
    b *= chan_scale(i1);
    c *= chan_scale(i2);
    d *= chan_scale(i3);

    // Write-once output: TH_STORE_NT so it doesn't evict the input from L2.
    __builtin_nontemporal_store(a, &out[i0]);
    __builtin_nontemporal_store(b, &out[i1]);
    __builtin_nontemporal_store(c, &out[i2]);
    __builtin_nontemporal_store(d, &out[i3]);
}

// Generic fallback (only launched if sizes ever stop dividing exactly).
__global__ __launch_bounds__(256) void
channel_scale_b128_guarded(const v4f* __restrict__ x, v4f* __restrict__ out,
                           unsigned n4, unsigned span) {
    const unsigned tid = blockIdx.x * blockDim.x + threadIdx.x;
#pragma unroll
    for (int j = 0; j < 4; ++j) {
        const unsigned i = tid + (unsigned)j * span;
        if (i < n4) {
            v4f v = x[i];
            v *= chan_scale(i);
            __builtin_nontemporal_store(v, &out[i]);
        }
    }
}

extern "C" void kernel_launch(void* const* d_in, const int* in_sizes, int n_in,
                              void* d_out, int out_size, void* d_ws, size_t ws_size,
                              hipStream_t stream) {
    (void)n_in; (void)d_ws; (void)ws_size; (void)out_size;

    const v4f* x = (const v4f*)d_in[0];
    v4f*       o = (v4f*)d_out;

    const unsigned n  = (unsigned)in_sizes[0];   // 33,554,432 elements
    const unsigned n4 = n >> 2;                  // 8,388,608 float4s (2^23)

    constexpr unsigned THREADS = 256;            // 8 wave32s per block
    constexpr unsigned UNROLL  = 4;              // float4s per thread
    constexpr unsigned TILE    = THREADS * UNROLL;

    if (n4 % TILE == 0u) {
        const unsigned blocks = n4 / TILE;       // 8192
        const unsigned span   = blocks * THREADS;
        channel_scale_b128_exact<<<blocks, THREADS, 0, stream>>>(x, o, span);
    } else {
        const unsigned total_threads = (n4 + UNROLL - 1) / UNROLL;
        const unsigned blocks = (total_threads + THREADS - 1) / THREADS;
        const unsigned span   = blocks * THREADS;
        channel_scale_b128_guarded<<<blocks, THREADS, 0, stream>>>(x, o, n4, span);
    }
}